// CompressedSensingInception_85942295593062
// MI455X (gfx1250) — compile-verified
//
#include <hip/hip_runtime.h>
#include <hip/hip_bf16.h>
#include <math.h>

// ---------------------------------------------------------------------------
// CompressedSensingInception for MI455X (gfx1250, wave32)
//   - prep_mat_kernel: build zero-padded copies of mat (guard-free hot loops)
//   - cs_solve_kernel: 100-iter FISTA loop on f32 WMMA (v_wmma_f32_16x16x4_f32)
//     24 WGs x 256 threads (8 waves), each owns 16 of 384 (b,c) rows.
//   - conv5pool / xpath2 / ywz: the tiny conv branches (scalar VALU).
// ---------------------------------------------------------------------------

typedef __attribute__((ext_vector_type(2))) float v2f;
typedef __attribute__((ext_vector_type(8))) float v8f;

constexpr int MDIM  = 5329;     // 73*73
constexpr int NDIM  = 81;       // 9*9
constexpr int NPAD  = 96;       // 6 WMMA n-tiles
constexpr int KPAD  = 5376;     // 4*1344 = 8*672 = 336*16 (zero-padded M/K)
constexpr int KB    = KPAD / 4; // 1344 k-blocks
constexpr int YS    = KPAD;     // Y row stride
constexpr int ITERS = 100;

// ================= one-shot: padded mat copies =============================
// matP[k][col]      : [5376][96] row-major (step-2 B: contiguous (kk,kk+1))
// matB[kb][col][s]  : [1344][96][4] k-interleaved (step-1 B: contiguous pair)
__global__ __launch_bounds__(256) void prep_mat_kernel(
    const float* __restrict__ mat, float* __restrict__ matP,
    float* __restrict__ matB)
{
  const int idx = blockIdx.x * 256 + threadIdx.x;
  if (idx >= KPAD * NPAD) return;
  const int col = idx % NPAD, k = idx / NPAD;
  const float v = (k < MDIM && col < NDIM) ? mat[(size_t)k * NDIM + col] : 0.f;
  matP[idx] = v;
  matB[(((size_t)(k >> 2) * NPAD) + col) * 4 + (k & 3)] = v;
}

// ======================= compressed-sensing FISTA loop =====================
__global__ __launch_bounds__(256) void cs_solve_kernel(
    const float* __restrict__ inp,   // [128,9,9,3]
    const float* __restrict__ matP,  // [5376,96] padded
    const float* __restrict__ matB,  // [1344,96,4] padded, k-interleaved
    const float* __restrict__ lam,   // [1]
    float* __restrict__ Y)           // [384, 5376] workspace
{
  __shared__ float lds_partial[8][16][NPAD];   // 48 KB
  __shared__ float lds_resid[16][NPAD];        // 6 KB
  __shared__ float lds_im[16][NPAD];           // 6 KB

  const int tid   = threadIdx.x;
  const int wave  = tid >> 5;
  const int lane  = tid & 31;
  const int l15   = lane & 15;
  const int khalf = lane >> 4;             // 0 or 1
  const int bc0   = blockIdx.x * 16;       // 16 (b,c) rows per WG
  const float thr = lam[0];                // mu == 1

  // Zero own Y rows (incl. padding; padding stays 0 forever).
  for (int e = tid; e < 16 * YS; e += 256)
    Y[(size_t)(bc0 + e / YS) * YS + (e % YS)] = 0.0f;

  // Hoist im tile into LDS (iteration-invariant). Padding cols -> 0.
  for (int e = tid; e < 16 * NPAD; e += 256) {
    const int row = e / NPAD, col = e % NPAD;
    float v = 0.f;
    if (col < NDIM) {
      const int bc = bc0 + row, b = bc / 3, c = bc % 3;
      v = inp[((b * 9 + col / 9) * 9 + (col % 9)) * 3 + c];
    }
    lds_im[row][col] = v;
  }
  __syncthreads();

  const int kb0  = wave * (KB / 8);        // 168 k-blocks per wave (step 1)
  const int tbeg = wave * 42;              // 42 m-tiles per wave (step 2)
  const float* __restrict__ Yrow = &Y[(size_t)(bc0 + l15) * YS];

  for (int it = 0; it < ITERS; ++it) {
    // ---- step 1: partial (Y_tile[16 x K]) * (mat[K x 96]), guard-free ----
    v8f acc[6];
    #pragma unroll
    for (int t = 0; t < 6; ++t)
      acc[t] = (v8f){0.f,0.f,0.f,0.f,0.f,0.f,0.f,0.f};

    #pragma unroll 2
    for (int kb = kb0; kb < kb0 + KB / 8; ++kb) {
      const int kk = (kb << 2) + 2 * khalf;
      const v2f a = *(const v2f*)(Yrow + kk);
      const float* __restrict__ bp = &matB[(size_t)kb * (NPAD * 4) + 2 * khalf];
      __builtin_prefetch(&matB[(size_t)(kb + 8) * (NPAD * 4)], 0, 1);
      #pragma unroll
      for (int t = 0; t < 6; ++t) {
        const v2f b = *(const v2f*)(bp + (t * 16 + l15) * 4);
        acc[t] = __builtin_amdgcn_wmma_f32_16x16x4_f32(
            false, a, false, b, (short)0, acc[t], false, false);
      }
    }
    // stage partials (C layout: vgpr r -> row = r + 8*khalf, col = 16t + l15)
    #pragma unroll
    for (int t = 0; t < 6; ++t)
      #pragma unroll
      for (int r = 0; r < 8; ++r)
        lds_partial[wave][r + 8 * khalf][t * 16 + l15] = acc[t][r];
    __syncthreads();

    // ---- cross-wave reduce + resid = im - Y*mat (pad cols stay 0) ----
    for (int e = tid; e < 16 * NPAD; e += 256) {
      const int row = e / NPAD, col = e % NPAD;
      float s = 0.f;
      #pragma unroll
      for (int w = 0; w < 8; ++w) s += lds_partial[w][row][col];
      lds_resid[row][col] = lds_im[row][col] - s;
    }
    __syncthreads();

    // ---- step 2: re = resid[16x84] * matP^T, dual m-tile, guard-free ----
    for (int tp = 0; tp < 21; ++tp) {
      const int m0 = ((tbeg + 2 * tp) << 4) + l15;     // tile t   column
      const float* __restrict__ mp0 = &matP[(size_t)m0 * NPAD];
      const float* __restrict__ mp1 = mp0 + 16 * NPAD; // tile t+1 column
      v8f acc2 = (v8f){0.f,0.f,0.f,0.f,0.f,0.f,0.f,0.f};
      v8f acc3 = (v8f){0.f,0.f,0.f,0.f,0.f,0.f,0.f,0.f};
      #pragma unroll
      for (int k0 = 0; k0 < 84; k0 += 4) {
        const int kk = k0 + 2 * khalf;
        const v2f a  = *(const v2f*)&lds_resid[l15][kk];
        const v2f b0 = *(const v2f*)(mp0 + kk);
        const v2f b1 = *(const v2f*)(mp1 + kk);
        acc2 = __builtin_amdgcn_wmma_f32_16x16x4_f32(
            false, a, false, b0, (short)0, acc2, false, false);
        acc3 = __builtin_amdgcn_wmma_f32_16x16x4_f32(
            false, a, false, b1, (short)0, acc3, false, false);
      }
      // soft-threshold update; padded m-columns compute/write exact zeros
      #pragma unroll
      for (int r = 0; r < 8; ++r) {
        const size_t y0 = (size_t)(bc0 + r + 8 * khalf) * YS + m0;
        const float w0 = Y[y0] + acc2[r];
        Y[y0] = fmaxf(w0 - thr, 0.f) - fmaxf(-w0 - thr, 0.f);
        const float w1 = Y[y0 + 16] + acc3[r];
        Y[y0 + 16] = fmaxf(w1 - thr, 0.f) - fmaxf(-w1 - thr, 0.f);
      }
    }
    __threadfence_block();
    __syncthreads();
  }
}

// =================== x branch: BN + conv5x5 + maxpool4 =====================
__global__ __launch_bounds__(256) void conv5pool_kernel(
    const float* __restrict__ Y, const float* __restrict__ bn_x,
    const float* __restrict__ w5, const float* __restrict__ b5,
    float* __restrict__ P)                       // [128,18,18,8]
{
  const int idx = blockIdx.x * 256 + threadIdx.x;
  if (idx >= 128 * 18 * 18) return;
  const int oj = idx % 18, oi = (idx / 18) % 18, b = idx / (18 * 18);

  float sc[3], sh[3];
  #pragma unroll
  for (int c = 0; c < 3; ++c) {
    const float g = bn_x[c], bb = bn_x[3 + c], m = bn_x[6 + c], v = bn_x[9 + c];
    sc[c] = g * rsqrtf(v + 1e-3f);
    sh[c] = bb - m * sc[c];
  }
  float mx[8];
  #pragma unroll
  for (int o = 0; o < 8; ++o) mx[o] = -__builtin_inff();

  for (int wi = 0; wi < 4; ++wi) {
    const int ci = 4 * oi - 1 + wi;              // SAME pool pad_lo = 1
    if (ci < 0 || ci >= 70) continue;
    for (int wj = 0; wj < 4; ++wj) {
      const int cj = 4 * oj - 1 + wj;
      if (cj < 0 || cj >= 70) continue;
      float acc[8];
      #pragma unroll
      for (int o = 0; o < 8; ++o) acc[o] = b5[o];
      for (int di = 0; di < 5; ++di) {
        const int pi = ci + di;                  // padded 0..73
        const int fi = (pi == 0) ? 1 : pi - 1;   // reflect pad (1,0)
        for (int dj = 0; dj < 5; ++dj) {
          const int pj = cj + dj;
          const int fj = (pj == 0) ? 1 : pj - 1;
          #pragma unroll
          for (int c = 0; c < 3; ++c) {
            const float xv =
                Y[(size_t)(b * 3 + c) * YS + fi * 73 + fj] * sc[c] + sh[c];
            const float* wp = &w5[((di * 5 + dj) * 3 + c) * 8];
            #pragma unroll
            for (int o = 0; o < 8; ++o) acc[o] = fmaf(xv, wp[o], acc[o]);
          }
        }
      }
      #pragma unroll
      for (int o = 0; o < 8; ++o) mx[o] = fmaxf(mx[o], acc[o]);
    }
  }
  float* Pp = &P[(size_t)idx * 8];
  #pragma unroll
  for (int o = 0; o < 8; ++o) Pp[o] = mx[o];
}

// =================== x branch tail: relu(1x1 conv) + maxpool2 ==============
__global__ __launch_bounds__(256) void xpath2_kernel(
    const float* __restrict__ P, const float* __restrict__ w12,
    const float* __restrict__ b12, float* __restrict__ out)
{
  const int idx = blockIdx.x * 256 + threadIdx.x;
  if (idx >= 128 * 81) return;
  const int j = idx % 9, i = (idx / 9) % 9, b = idx / 81;
  #pragma unroll
  for (int o = 0; o < 2; ++o) {
    float mx = -__builtin_inff();
    for (int wi = 0; wi < 2; ++wi)
      for (int wj = 0; wj < 2; ++wj) {
        const float* p =
            &P[(((size_t)b * 18 + (2 * i + wi)) * 18 + (2 * j + wj)) * 8];
        float v = b12[o];
        #pragma unroll
        for (int ic = 0; ic < 8; ++ic) v = fmaf(p[ic], w12[ic * 2 + o], v);
        mx = fmaxf(mx, fmaxf(v, 0.f));
      }
    out[(size_t)idx * 5 + 1 + o] = mx;
  }
}

// =================== w / y / z branches (one image per block) ==============
__global__ __launch_bounds__(128) void ywz_kernel(
    const float* __restrict__ inp,
    const float* __restrict__ w_y1, const float* __restrict__ b_y1,
    const float* __restrict__ w_y7, const float* __restrict__ b_y7,
    const float* __restrict__ bn_y,
    const float* __restrict__ w_w1, const float* __restrict__ b_w1,
    const float* __restrict__ w_d1, const float* __restrict__ bn_d1,
    const float* __restrict__ w_d2, const float* __restrict__ bn_d2,
    const float* __restrict__ w_u1, const float* __restrict__ bn_u1,
    const float* __restrict__ w_u2, const float* __restrict__ bn_u2,
    float* __restrict__ out)
{
  __shared__ float y1[9][9];
  __shared__ float z1s[3][3][12];
  __shared__ float z2s[24];
  __shared__ float zc[3][3][24];

  const int b = blockIdx.x, tid = threadIdx.x;
  const float* ib = &inp[(size_t)b * 81 * 3];
  float* ob = &out[(size_t)b * 81 * 5];

  if (tid < 81) {   // w path + y1
    const float* px = &ib[tid * 3];
    float wv = b_w1[0], yv = b_y1[0];
    #pragma unroll
    for (int c = 0; c < 3; ++c) {
      wv = fmaf(px[c], w_w1[c], wv);
      yv = fmaf(px[c], w_y1[c], yv);
    }
    ob[tid * 5 + 0] = fmaxf(wv, 0.f);
    y1[tid / 9][tid % 9] = fmaxf(yv, 0.f);
  }
  if (tid < 108) {  // z1 = lrelu(bn(conv3x3 s3))  [3,3,12]
    const int oc = tid % 12, oj = (tid / 12) % 3, oi = tid / 36;
    float acc = 0.f;
    for (int ki = 0; ki < 3; ++ki)
      for (int kj = 0; kj < 3; ++kj) {
        const float* px = &ib[((3 * oi + ki) * 9 + (3 * oj + kj)) * 3];
        #pragma unroll
        for (int c = 0; c < 3; ++c)
          acc = fmaf(px[c], w_d1[((ki * 3 + kj) * 3 + c) * 12 + oc], acc);
      }
    const float z = (acc - bn_d1[24 + oc]) * rsqrtf(bn_d1[36 + oc] + 1e-3f) *
                        bn_d1[oc] + bn_d1[12 + oc];
    z1s[oi][oj][oc] = (z > 0.f) ? z : 0.3f * z;
  }
  __syncthreads();

  if (tid < 24) {   // z2 = lrelu(bn(conv3x3 s3))  [1,1,24]
    float acc = 0.f;
    for (int ki = 0; ki < 3; ++ki)
      for (int kj = 0; kj < 3; ++kj)
        #pragma unroll
        for (int ic = 0; ic < 12; ++ic)
          acc = fmaf(z1s[ki][kj][ic], w_d2[((ki * 3 + kj) * 12 + ic) * 24 + tid], acc);
    const float z = (acc - bn_d2[48 + tid]) * rsqrtf(bn_d2[72 + tid] + 1e-3f) *
                        bn_d2[tid] + bn_d2[24 + tid];
    z2s[tid] = (z > 0.f) ? z : 0.3f * z;
  }
  if (tid < 81) {   // y = bn(conv7x7 SAME(y1))
    const int i = tid / 9, j = tid % 9;
    float acc = b_y7[0];
    for (int di = 0; di < 7; ++di) {
      const int si = i + di - 3;
      if (si < 0 || si > 8) continue;
      for (int dj = 0; dj < 7; ++dj) {
        const int sj = j + dj - 3;
        if (sj < 0 || sj > 8) continue;
        acc = fmaf(y1[si][sj], w_y7[di * 7 + dj], acc);
      }
    }
    ob[tid * 5 + 3] =
        (acc - bn_y[2]) * rsqrtf(bn_y[3] + 1e-3f) * bn_y[0] + bn_y[1];
  }
  __syncthreads();

  if (tid < 108) {  // u1 = relu(bn(convT(z2))) ; concat with z1 -> zc[3,3,24]
    const int oc = tid % 12, oj = (tid / 12) % 3, oi = tid / 36;
    float acc = 0.f;
    #pragma unroll
    for (int ic = 0; ic < 24; ++ic)
      acc = fmaf(z2s[ic], w_u1[((oi * 3 + oj) * 24 + ic) * 12 + oc], acc);
    const float z = (acc - bn_u1[24 + oc]) * rsqrtf(bn_u1[36 + oc] + 1e-3f) *
                        bn_u1[oc] + bn_u1[12 + oc];
    zc[oi][oj][oc] = fmaxf(z, 0.f);
    zc[oi][oj][12 + oc] = z1s[oi][oj][oc];
  }
  __syncthreads();

  if (tid < 81) {   // z = relu(bn(convT(zc))) -> [9,9,1]
    const int i = tid / 9, j = tid % 9;
    const int ii = i / 3, ki = i % 3, jj = j / 3, kj = j % 3;
    float acc = 0.f;
    #pragma unroll
    for (int ic = 0; ic < 24; ++ic)
      acc = fmaf(zc[ii][jj][ic], w_u2[(ki * 3 + kj) * 24 + ic], acc);
    const float z =
        (acc - bn_u2[2]) * rsqrtf(bn_u2[3] + 1e-3f) * bn_u2[0] + bn_u2[1];
    ob[tid * 5 + 4] = fmaxf(z, 0.f);
  }
}

// ============================== launcher ===================================
extern "C" void kernel_launch(void* const* d_in, const int* in_sizes, int n_in,
                              void* d_out, int out_size, void* d_ws, size_t ws_size,
                              hipStream_t stream)
{
  const float* inp   = (const float*)d_in[0];
  const float* mat   = (const float*)d_in[1];
  const float* lam   = (const float*)d_in[2];
  const float* bn_x  = (const float*)d_in[3];
  const float* w5    = (const float*)d_in[4];
  const float* b5    = (const float*)d_in[5];
  const float* w12   = (const float*)d_in[6];
  const float* b12   = (const float*)d_in[7];
  const float* w_y1  = (const float*)d_in[8];
  const float* b_y1  = (const float*)d_in[9];
  const float* w_y7  = (const float*)d_in[10];
  const float* b_y7  = (const float*)d_in[11];
  const float* bn_y  = (const float*)d_in[12];
  const float* w_w1  = (const float*)d_in[13];
  const float* b_w1  = (const float*)d_in[14];
  const float* w_d1  = (const float*)d_in[15];
  const float* bn_d1 = (const float*)d_in[16];
  const float* w_d2  = (const float*)d_in[17];
  const float* bn_d2 = (const float*)d_in[18];
  const float* w_u1  = (const float*)d_in[19];
  const float* bn_u1 = (const float*)d_in[20];
  const float* w_u2  = (const float*)d_in[21];
  const float* bn_u2 = (const float*)d_in[22];
  float* out = (float*)d_out;

  float* Y    = (float*)d_ws;                        // 384*5376 f32 (~8.3 MB)
  float* matP = Y    + (size_t)384 * YS;             // 5376*96    (~2.1 MB)
  float* matB = matP + (size_t)KPAD * NPAD;          // 1344*96*4  (~2.1 MB)
  float* P    = matB + (size_t)KB * NPAD * 4;        // 128*18*18*8 (~1.3 MB)

  prep_mat_kernel<<<(KPAD * NPAD + 255) / 256, 256, 0, stream>>>(mat, matP, matB);
  cs_solve_kernel<<<24, 256, 0, stream>>>(inp, matP, matB, lam, Y);
  conv5pool_kernel<<<(128 * 18 * 18 + 255) / 256, 256, 0, stream>>>(Y, bn_x, w5, b5, P);
  xpath2_kernel<<<(128 * 81 + 255) / 256, 256, 0, stream>>>(P, w12, b12, out);
  ywz_kernel<<<128, 128, 0, stream>>>(inp, w_y1, b_y1, w_y7, b_y7, bn_y,
                                      w_w1, b_w1, w_d1, bn_d1, w_d2, bn_d2,
                                      w_u1, bn_u1, w_u2, bn_u2, out);
}